// ViTPatchTokenizer_50629074485716
// MI455X (gfx1250) — compile-verified
//
#include <hip/hip_runtime.h>
#include <hip/hip_bf16.h>

// ViT patch tokenizer front-end for img (16,3,512,512) f32, PATCH=16.
// Outputs (flat, concatenated, all as float32 in d_out):
//   fV  [B*H*W, 3]  : NCHW -> NHWC permute            (12,582,912 floats)
//   seg [B*H*W]     : patch id per pixel               ( 4,194,304)
//   byx [3, B*H*W]  : (b,y,x) per pixel                (12,582,912)
//   bb  [4, nV]     : per-patch bbox (analytic)        (     65,536)
// Pure bandwidth problem (~160 MiB @ 23.3 TB/s ~= 7us). Strategy: coalesced
// B128 channel loads, register interleave, and CDNA5 async LDS->global B128
// stores for the stride-3 fV stream (ASYNCcnt path).

#define BLOCK 256

typedef float v4f __attribute__((ext_vector_type(4)));
typedef int   v4i __attribute__((vector_size(16)));   // matches builtin param

typedef __attribute__((address_space(1))) v4i* g_v4i_ptr;  // __device__ int4*
typedef __attribute__((address_space(3))) v4i* l_v4i_ptr;  // __shared__ int4*

#if defined(__gfx1250__) &&                                                    \
    __has_builtin(__builtin_amdgcn_global_store_async_from_lds_b128) &&        \
    __has_builtin(__builtin_amdgcn_s_wait_asynccnt)
#define USE_ASYNC_LDS 1
#else
#define USE_ASYNC_LDS 0
#endif

__global__ __launch_bounds__(BLOCK) void
vit_tok_main(const float* __restrict__ img,
             float* __restrict__ out_fV,
             float* __restrict__ out_seg,
             float* __restrict__ out_byx) {
  constexpr int HW = 512 * 512;        // 262144 = 2^18
  constexpr int N  = 16 * HW;          // total pixels
  const int tid  = threadIdx.x;
  const int g    = blockIdx.x * BLOCK + tid;  // pixel-group id (4 px/thread)
  const int pix0 = g << 2;                    // first pixel of group
  const int b    = pix0 >> 18;
  const int rem  = pix0 & (HW - 1);
  const int y    = rem >> 9;
  const int x    = rem & 511;                 // multiple of 4

  // Coalesced 16B loads, one per channel plane.
  const float* base = img + (size_t)(b * 3) * HW + (y << 9) + x;
  v4f L0 = *(const v4f*)(base + 0 * HW);
  v4f L1 = *(const v4f*)(base + 1 * HW);
  v4f L2 = *(const v4f*)(base + 2 * HW);

  // Interleave: out[12g + 3k + c] = L[c][k]
  v4f s0 = {L0.x, L1.x, L2.x, L0.y};
  v4f s1 = {L1.y, L2.y, L0.z, L1.z};
  v4f s2 = {L2.z, L0.w, L1.w, L2.w};

#if USE_ASYNC_LDS
  // Stage block's 12KB fV tile in LDS, drain with contiguous async B128 stores.
  __shared__ __align__(16) float lds[BLOCK * 12];
  {
    v4f* lp = (v4f*)&lds[tid * 12];   // 48B per thread, 16B aligned
    lp[0] = s0; lp[1] = s1; lp[2] = s2;
  }
  __syncthreads();
  {
    char* gbase = (char*)out_fV + (size_t)blockIdx.x * (BLOCK * 12 * 4);
    char* lbase = (char*)lds;
#pragma unroll
    for (int k = 0; k < 3; ++k) {
      const int chunk = tid + k * BLOCK;  // 768 x 16B chunks
      __builtin_amdgcn_global_store_async_from_lds_b128(
          (g_v4i_ptr)(gbase + chunk * 16),
          (l_v4i_ptr)(lbase + chunk * 16),
          0, 0);
    }
  }
#else
  {
    v4f* o = (v4f*)(out_fV + (size_t)g * 12);
    o[0] = s0; o[1] = s1; o[2] = s2;
  }
#endif

  // seg: all 4 pixels of the group share one patch (x%16 <= 12).
  const float segf = (float)((b << 10) + ((y >> 4) << 5) + (x >> 4));
  *(v4f*)(out_seg + pix0) = (v4f){segf, segf, segf, segf};

  // byx planes: [0]=b, [1]=y, [2]=x  (coalesced B128 stores)
  const float bf = (float)b, yf = (float)y, xf = (float)x;
  *(v4f*)(out_byx + pix0)         = (v4f){bf, bf, bf, bf};
  *(v4f*)(out_byx + N + pix0)     = (v4f){yf, yf, yf, yf};
  *(v4f*)(out_byx + 2 * N + pix0) = (v4f){xf, xf + 1.f, xf + 2.f, xf + 3.f};

#if USE_ASYNC_LDS
  __builtin_amdgcn_s_wait_asynccnt(0);
#endif
}

// bb is analytic: patches are full 16x16 tiles, so min/max are closed-form.
__global__ __launch_bounds__(BLOCK) void vit_tok_bb(float* __restrict__ out_bb) {
  constexpr int nV = 16 * 32 * 32;  // 16384
  const int s = blockIdx.x * BLOCK + threadIdx.x;
  const int rem = s & 1023;         // within-batch patch id
  const float ymin = (float)((rem >> 5) << 4);
  const float xmin = (float)((rem & 31) << 4);
  out_bb[0 * nV + s] = ymin;
  out_bb[1 * nV + s] = xmin;
  out_bb[2 * nV + s] = ymin + 15.f;
  out_bb[3 * nV + s] = xmin + 15.f;
}

extern "C" void kernel_launch(void* const* d_in, const int* in_sizes, int n_in,
                              void* d_out, int out_size, void* d_ws, size_t ws_size,
                              hipStream_t stream) {
  (void)in_sizes; (void)n_in; (void)d_ws; (void)ws_size; (void)out_size;
  const float* img = (const float*)d_in[0];
  float* out = (float*)d_out;

  constexpr int N  = 16 * 512 * 512;      // 4,194,304 pixels
  constexpr int nV = 16 * 32 * 32;        // 16,384 patches

  float* out_fV  = out;                   // N*3
  float* out_seg = out + (size_t)N * 3;   // N
  float* out_byx = out_seg + N;           // 3*N
  float* out_bb  = out_byx + (size_t)N * 3;  // 4*nV

  const int groups = N / 4;               // 4 pixels per thread
  vit_tok_main<<<groups / BLOCK, BLOCK, 0, stream>>>(img, out_fV, out_seg, out_byx);
  vit_tok_bb<<<nV / BLOCK, BLOCK, 0, stream>>>(out_bb);
}